// QuantumTransformerBlock_65481071396699
// MI455X (gfx1250) — compile-verified
//
#include <hip/hip_runtime.h>
#include <hip/hip_bf16.h>
#include <math.h>

// 10-qubit statevector simulator, one wave32 per sample.
// State = 32x32 complex matrix in LDS (rows = wires 0-4, cols = wires 5-9).
// Gate blocks applied as 32x32x32 complex GEMMs via V_WMMA_F32_16X16X4_F32.

typedef __attribute__((ext_vector_type(2))) float v2f;
typedef __attribute__((ext_vector_type(8))) float v8f;

#define LD 33          // padded LDS leading dim for 32x32 planes
#define NSAMP 4096     // 8 * 512
#define WAVES_PER_BLK 2

__device__ __forceinline__ v8f wmma4(v2f a, v2f b, v8f c) {
  // D = A(16x4) * B(4x16) + C ; f32 WMMA, 8-arg form per CDNA5 probe.
  return __builtin_amdgcn_wmma_f32_16x16x4_f32(
      false, a, false, b, (short)0, c, false, false);
}

// 32x32 GEMM on one wave: D = A * B, optionally complex A/B, optionally
// loading B transposed (B[k][n] = Bsrc[n][k]).  A-operand layout: lane l
// holds rows M = l&15, K pair = Kt + 2*(l>>4).  B-operand: lane l holds
// col N = l&15, K pair = Kt + 2*(l>>4).  C/D: lane l holds col N = l&15,
// rows Mt + 8*(l>>4) + j in element j.
template <bool ACPLX, bool BCPLX, bool BTRANS>
__device__ __forceinline__ void gemm32(const float* __restrict__ Are,
                                       const float* __restrict__ Aim, int lda,
                                       const float* __restrict__ Bre,
                                       const float* __restrict__ Bim, int ldb,
                                       float* __restrict__ Dre,
                                       float* __restrict__ Dim, int ldd,
                                       int lane) {
  const int lhalf = lane >> 4;
  const int lmod  = lane & 15;
#pragma unroll
  for (int Mt = 0; Mt < 32; Mt += 16) {
#pragma unroll
    for (int Nt = 0; Nt < 32; Nt += 16) {
      v8f cr = {};
      v8f ci = {};
#pragma unroll
      for (int Kt = 0; Kt < 32; Kt += 4) {
        const int m  = Mt + lmod;
        const int n  = Nt + lmod;
        const int k2 = Kt + (lhalf << 1);
        v2f ar, ai, br, bi;
        ar.x = Are[m * lda + k2];
        ar.y = Are[m * lda + k2 + 1];
        if (ACPLX) {
          ai.x = Aim[m * lda + k2];
          ai.y = Aim[m * lda + k2 + 1];
        }
        if (BTRANS) {
          br.x = Bre[n * ldb + k2];
          br.y = Bre[n * ldb + k2 + 1];
          if (BCPLX) {
            bi.x = Bim[n * ldb + k2];
            bi.y = Bim[n * ldb + k2 + 1];
          }
        } else {
          br.x = Bre[k2 * ldb + n];
          br.y = Bre[(k2 + 1) * ldb + n];
          if (BCPLX) {
            bi.x = Bim[k2 * ldb + n];
            bi.y = Bim[(k2 + 1) * ldb + n];
          }
        }
        cr = wmma4(ar, br, cr);
        if (ACPLX && BCPLX) {
          v2f nai;                       // f32 WMMA has no A-negate; do it in VALU
          nai.x = -ai.x;
          nai.y = -ai.y;
          cr = wmma4(nai, bi, cr);       // re -= Aim*Bim
          ci = wmma4(ar, bi, ci);        // im += Are*Bim
          ci = wmma4(ai, br, ci);        // im += Aim*Bre
        } else if (ACPLX) {              // B real
          ci = wmma4(ai, br, ci);
        } else if (BCPLX) {              // A real
          ci = wmma4(ar, bi, ci);
        }
      }
      const int n  = Nt + lmod;
      const int mb = Mt + (lhalf << 3);
#pragma unroll
      for (int j = 0; j < 8; ++j) {
        Dre[(mb + j) * ldd + n] = cr[j];
        Dim[(mb + j) * ldd + n] = ci[j];
      }
    }
  }
}

// Prep: fuse RZ(p2)*RY(p1)*RX(p0) per (layer,wire), then Kronecker into two
// 32x32 complex matrices per layer: mat 0 = wires 0-4 (rows, left-mult),
// mat 1 = wires 5-9 (cols, right-mult).  ws layout:
// ws[((layer*2+mat)*2+comp)*1024 + r*32 + q], comp 0=re 1=im.
__global__ void qprep_kernel(const float* __restrict__ p,
                             float* __restrict__ ws) {
  int idx = blockIdx.x * blockDim.x + threadIdx.x;
  if (idx >= 4096) return;
  const int layer = idx >> 11;
  const int mat   = (idx >> 10) & 1;
  const int e     = idx & 1023;
  const int r     = e >> 5;
  const int q     = e & 31;
  float pr = 1.0f, pi = 0.0f;
#pragma unroll
  for (int j = 0; j < 5; ++j) {
    const int wire = mat * 5 + j;
    const float a  = p[(layer * 10 + wire) * 3 + 0];
    const float b  = p[(layer * 10 + wire) * 3 + 1];
    const float c  = p[(layer * 10 + wire) * 3 + 2];
    const float cx = cosf(0.5f * a), sx = sinf(0.5f * a);
    const float cy = cosf(0.5f * b), sy = sinf(0.5f * b);
    const float cz = cosf(0.5f * c), sz = sinf(0.5f * c);
    const int ab = (r >> (4 - j)) & 1;   // row bit for this wire
    const int bb = (q >> (4 - j)) & 1;   // col bit
    // M = RY*RX entry (ab,bb)
    float mr, mi;
    if (ab == 0 && bb == 0)      { mr =  cy * cx; mi =  sy * sx; }
    else if (ab == 0)            { mr = -sy * cx; mi = -cy * sx; }
    else if (bb == 0)            { mr =  sy * cx; mi = -cy * sx; }
    else                         { mr =  cy * cx; mi = -sy * sx; }
    // RZ phase: row0 * (cz - i sz), row1 * (cz + i sz)
    const float phr = cz;
    const float phi = ab ? sz : -sz;
    const float gr = phr * mr - phi * mi;
    const float gi = phr * mi + phi * mr;
    const float nr = pr * gr - pi * gi;
    const float ni = pr * gi + pi * gr;
    pr = nr;
    pi = ni;
  }
  const int base = (layer * 2 + mat) * 2;
  ws[(base + 0) * 1024 + e] = pr;
  ws[(base + 1) * 1024 + e] = pi;
}

__global__ __launch_bounds__(64) void qsim_kernel(const float* __restrict__ x,
                                                  const float* __restrict__ ws,
                                                  float* __restrict__ out) {
  __shared__ float smem[WAVES_PER_BLK][5][32 * LD];
  const int lane   = threadIdx.x & 31;
  const int wid    = threadIdx.x >> 5;
  const int sample = blockIdx.x * WAVES_PER_BLK + wid;

  float* b0re = smem[wid][0];
  float* b0im = smem[wid][1];
  float* b1re = smem[wid][2];
  float* b1im = smem[wid][3];
  float* kmat = smem[wid][4];

  // |0...0>
#pragma unroll
  for (int t = 0; t < 32; ++t) {
    b0re[t * LD + lane] = 0.0f;
    b0im[t * LD + lane] = 0.0f;
  }
  if (lane == 0) b0re[0] = 1.0f;
  __syncthreads();

  float* cre = b0re;
  float* cim = b0im;
  float* nre = b1re;
  float* nim = b1im;

  for (int l = 0; l < 2; ++l) {
    // ---- RY(x) kron for wires 0-4 (row matrix), real ----
    {
      float c[5], s[5];
#pragma unroll
      for (int j = 0; j < 5; ++j) {
        const float xw = x[sample * 10 + j];
        c[j] = cosf(0.5f * xw);
        s[j] = sinf(0.5f * xw);
      }
#pragma unroll
      for (int t = 0; t < 32; ++t) {
        float prod = 1.0f;
#pragma unroll
        for (int j = 0; j < 5; ++j) {
          const int ab = (t >> (4 - j)) & 1;
          const int bb = (lane >> (4 - j)) & 1;
          prod *= (ab == bb) ? c[j] : (ab ? s[j] : -s[j]);
        }
        kmat[t * LD + lane] = prod;
      }
    }
    __syncthreads();
    gemm32<false, true, false>(kmat, nullptr, LD, cre, cim, LD, nre, nim, LD,
                               lane);   // S = Khi * S
    __syncthreads();
    { float* t0 = cre; cre = nre; nre = t0; t0 = cim; cim = nim; nim = t0; }

    // ---- RY(x) kron for wires 5-9 (col matrix), real ----
    {
      float c[5], s[5];
#pragma unroll
      for (int j = 0; j < 5; ++j) {
        const float xw = x[sample * 10 + 5 + j];
        c[j] = cosf(0.5f * xw);
        s[j] = sinf(0.5f * xw);
      }
#pragma unroll
      for (int t = 0; t < 32; ++t) {
        float prod = 1.0f;
#pragma unroll
        for (int j = 0; j < 5; ++j) {
          const int ab = (t >> (4 - j)) & 1;
          const int bb = (lane >> (4 - j)) & 1;
          prod *= (ab == bb) ? c[j] : (ab ? s[j] : -s[j]);
        }
        kmat[t * LD + lane] = prod;
      }
    }
    __syncthreads();
    gemm32<true, false, true>(cre, cim, LD, kmat, nullptr, LD, nre, nim, LD,
                              lane);    // S = S * Klo^T
    __syncthreads();
    { float* t0 = cre; cre = nre; nre = t0; t0 = cim; cim = nim; nim = t0; }

    // ---- CNOT ring: single composed basis permutation (gather) ----
#pragma unroll
    for (int t = 0; t < 32; ++t) {
      const unsigned y = (unsigned)(t * 32 + lane);
      unsigned v = y;                      // composed gather index
      v ^= ((v >> 0) & 1u) << 9;           // CNOT(w9 -> w0)
      v ^= ((v >> 1) & 1u) << 0;           // CNOT(w8 -> w9)
      v ^= ((v >> 2) & 1u) << 1;
      v ^= ((v >> 3) & 1u) << 2;
      v ^= ((v >> 4) & 1u) << 3;
      v ^= ((v >> 5) & 1u) << 4;
      v ^= ((v >> 6) & 1u) << 5;
      v ^= ((v >> 7) & 1u) << 6;
      v ^= ((v >> 8) & 1u) << 7;
      v ^= ((v >> 9) & 1u) << 8;           // CNOT(w0 -> w1)
      const int yi = (int)((y >> 5) * LD + (y & 31));
      const int xi = (int)((v >> 5) * LD + (v & 31));
      nre[yi] = cre[xi];
      nim[yi] = cim[xi];
    }
    __syncthreads();
    { float* t0 = cre; cre = nre; nre = t0; t0 = cim; cim = nim; nim = t0; }

    // ---- fused param gates, wires 0-4: S = Uhi * S (complex*complex) ----
    const float* gre = ws + ((l * 2 + 0) * 2 + 0) * 1024;
    const float* gim = ws + ((l * 2 + 0) * 2 + 1) * 1024;
    gemm32<true, true, false>(gre, gim, 32, cre, cim, LD, nre, nim, LD, lane);
    __syncthreads();
    { float* t0 = cre; cre = nre; nre = t0; t0 = cim; cim = nim; nim = t0; }

    // ---- fused param gates, wires 5-9: S = S * Ulo^T ----
    const float* hre = ws + ((l * 2 + 1) * 2 + 0) * 1024;
    const float* him = ws + ((l * 2 + 1) * 2 + 1) * 1024;
    gemm32<true, true, true>(cre, cim, LD, hre, him, 32, nre, nim, LD, lane);
    __syncthreads();
    { float* t0 = cre; cre = nre; nre = t0; t0 = cim; cim = nim; nim = t0; }
  }

  // ---- <Z_w> = sum_i p_i * (1 - 2*bit(i, 9-w)) ----
  float acc[10];
#pragma unroll
  for (int w = 0; w < 10; ++w) acc[w] = 0.0f;
#pragma unroll
  for (int t = 0; t < 32; ++t) {
    const int i = t * 32 + lane;
    const float re = cre[t * LD + lane];
    const float im = cim[t * LD + lane];
    const float pb = re * re + im * im;
#pragma unroll
    for (int w = 0; w < 10; ++w) {
      acc[w] += ((i >> (9 - w)) & 1) ? -pb : pb;
    }
  }
#pragma unroll
  for (int w = 0; w < 10; ++w) {
#pragma unroll
    for (int off = 16; off > 0; off >>= 1) {
      acc[w] += __shfl_xor(acc[w], off, 32);
    }
  }
  if (lane == 0) {
#pragma unroll
    for (int w = 0; w < 10; ++w) out[sample * 10 + w] = acc[w];
  }
}

extern "C" void kernel_launch(void* const* d_in, const int* in_sizes, int n_in,
                              void* d_out, int out_size, void* d_ws,
                              size_t ws_size, hipStream_t stream) {
  const float* x = (const float*)d_in[0];   // [8,512,10] f32
  const float* p = (const float*)d_in[1];   // [2,10,3]  f32
  float* out = (float*)d_out;               // [8,512,10] f32
  float* ws  = (float*)d_ws;                // 32KB: fused param matrices

  qprep_kernel<<<16, 256, 0, stream>>>(p, ws);
  qsim_kernel<<<NSAMP / WAVES_PER_BLK, 64, 0, stream>>>(x, ws, out);
}